// ChamferKLLoss_19121194402402
// MI455X (gfx1250) — compile-verified
//
#include <hip/hip_runtime.h>

typedef __attribute__((ext_vector_type(2))) float v2f;
typedef __attribute__((ext_vector_type(8))) float v8f;

#define NPTS 4096
#define BATCH 16

__global__ void ChamferKL_zero_kernel(float* out) {
    if (threadIdx.x == 0) out[0] = 0.0f;
}

// One wave per 16-row strip of X, sweeping all 4096 columns of Y in 16-wide
// tiles. P[M,N] = rx[M] + ry[N] - 2*x_M.y_N comes straight out of one WMMA:
//   A = (-2*X_tile) 16x4 f32 (K=3 padded to 4), B = Y_tile 4x16 f32,
//   C = rx[M] + ry[N]  ->  D = P tile.
// blockIdx.y selects the direction (rows=gts or rows=preds); P is symmetric
// under the swap, so each pass computes one of the two chamfer sums fully
// locally (row-min only), ending in a single atomicAdd per wave.
__global__ __launch_bounds__(256)
void ChamferKL_dir_kernel(const float* __restrict__ preds,
                          const float* __restrict__ gts,
                          float* __restrict__ out)
{
    const float* __restrict__ X = (blockIdx.y == 0) ? gts   : preds; // row side
    const float* __restrict__ Y = (blockIdx.y == 0) ? preds : gts;   // col side

    const int lane  = threadIdx.x & 31;
    const int wave  = threadIdx.x >> 5;
    const int strip = blockIdx.x * 8 + wave;     // 0..4095
    const int b     = strip >> 8;                // batch (256 strips per batch)
    const int rtile = strip & 255;               // 16-row tile within batch
    const int n_base = rtile * 16;
    const size_t bbase = (size_t)b * 3 * NPTS;
    const int half = lane >> 4;                  // 0: K=0,1   1: K=2,3(pad)
    const int l16  = lane & 15;

    // ---- A tile (rows of X, scaled by -2). ISA 32-bit A 16x4 layout:
    // lanes 0-15:  row=lane,    VGPR0=K0, VGPR1=K1
    // lanes 16-31: row=lane-16, VGPR0=K2, VGPR1=K3 (pad 0)
    const int arow = n_base + l16;
    {
        // branchless one-time setup
    }
    const float xa0 = X[bbase + 0 * NPTS + arow];
    const float xa1 = X[bbase + 1 * NPTS + arow];
    const float xa2 = X[bbase + 2 * NPTS + arow];
    v2f A;
    A.x = -2.0f * (half ? xa2 : xa0);
    A.y = half ? 0.0f : (-2.0f * xa1);

    // ---- rx for the 8 accumulator rows this lane covers: row = k + 8*half
    v8f rxv;
#pragma unroll
    for (int k = 0; k < 8; ++k) {
        const int r = n_base + k + 8 * half;
        const float c0 = X[bbase + 0 * NPTS + r];
        const float c1 = X[bbase + 1 * NPTS + r];
        const float c2 = X[bbase + 2 * NPTS + r];
        rxv[k] = fmaf(c0, c0, fmaf(c1, c1, c2 * c2));
    }

    v8f runmin;
#pragma unroll
    for (int k = 0; k < 8; ++k) runmin[k] = 3.0e38f;

    const float* __restrict__ y0p = Y + bbase;
    const float* __restrict__ y1p = Y + bbase + NPTS;
    const float* __restrict__ y2p = Y + bbase + 2 * NPTS;
    const float b1mask = half ? 0.0f : 1.0f;

#pragma unroll 4
    for (int mt = 0; mt < NPTS / 16; ++mt) {
        const int col = mt * 16 + l16;
        // Three independent cached loads; no cross-lane exchange needed for ry.
        const float y0 = y0p[col];
        const float y1 = y1p[col];
        const float y2 = y2p[col];

        // ---- B tile (cols of Y), branchless lane-half select:
        // lanes 0-15: (K0,K1)=(y0,y1) ; lanes 16-31: (K2,K3)=(y2,0)
        v2f Bv;
        Bv.x = half ? y2 : y0;       // v_cndmask
        Bv.y = y1 * b1mask;          // zero pad on upper half

        const float ry = fmaf(y0, y0, fmaf(y1, y1, y2 * y2));

        v8f c;
#pragma unroll
        for (int k = 0; k < 8; ++k) c[k] = rxv[k] + ry;

        // D = (-2X)*Y + (rx+ry) == P tile, directly from the matrix unit
        const v8f d = __builtin_amdgcn_wmma_f32_16x16x4_f32(
            /*neg_a=*/false, A, /*neg_b=*/false, Bv,
            /*c_mod=*/(short)0, c, /*reuse_a=*/false, /*reuse_b=*/false);

#pragma unroll
        for (int k = 0; k < 8; ++k) runmin[k] = fminf(runmin[k], d[k]);
    }

    // ---- row-min reduction: min over the 16 lanes within each half (xor
    // masks < 16 stay inside the half), sum the 8 row-mins, combine halves.
    float s = 0.0f;
#pragma unroll
    for (int k = 0; k < 8; ++k) {
        float m = runmin[k];
        m = fminf(m, __shfl_xor(m, 1, 32));
        m = fminf(m, __shfl_xor(m, 2, 32));
        m = fminf(m, __shfl_xor(m, 4, 32));
        m = fminf(m, __shfl_xor(m, 8, 32));
        s += m;                                  // replicated across half's lanes
    }
    s += __shfl_xor(s, 16, 32);                  // rows 0-7 (+) rows 8-15
    if (lane == 0) atomicAdd(out, s);
}

__global__ __launch_bounds__(256)
void ChamferKL_kl_kernel(const float* __restrict__ mu,
                         const float* __restrict__ logvar,
                         float* __restrict__ out, int n)
{
    __shared__ float red[256];
    float s = 0.0f;
    for (int i = threadIdx.x; i < n; i += 256) {
        const float lv = logvar[i];
        const float m  = mu[i];
        s += 1.0f + lv - m * m - expf(lv);
    }
    red[threadIdx.x] = s;
    __syncthreads();
    for (int off = 128; off > 0; off >>= 1) {
        if (threadIdx.x < off) red[threadIdx.x] += red[threadIdx.x + off];
        __syncthreads();
    }
    if (threadIdx.x == 0) atomicAdd(out, -0.5f * red[0]);
}

extern "C" void kernel_launch(void* const* d_in, const int* in_sizes, int n_in,
                              void* d_out, int out_size, void* d_ws, size_t ws_size,
                              hipStream_t stream) {
    const float* preds  = (const float*)d_in[0];  // [16,3,4096]
    const float* gts    = (const float*)d_in[1];  // [16,3,4096]
    const float* mu     = (const float*)d_in[2];  // [16,128]
    const float* logvar = (const float*)d_in[3];  // [16,128]
    float* out = (float*)d_out;                   // scalar f32

    ChamferKL_zero_kernel<<<1, 64, 0, stream>>>(out);

    // 4096 strips per direction / 8 waves per block = 512 blocks; y = direction
    dim3 grid(512, 2), block(256);
    ChamferKL_dir_kernel<<<grid, block, 0, stream>>>(preds, gts, out);

    ChamferKL_kl_kernel<<<1, 256, 0, stream>>>(mu, logvar, out, BATCH * 128);
}